// GCATLayer_2439541424731
// MI455X (gfx1250) — compile-verified
//
#include <hip/hip_runtime.h>
#include <hip/hip_bf16.h>

// Sizes fixed by the reference
#define NN 4000
#define LL 32
#define DD 128
#define EE 16000
#define MIN_VAL (-1e15f)

typedef __attribute__((ext_vector_type(2))) float v2f;
typedef __attribute__((ext_vector_type(8))) float v8f;

static __device__ __forceinline__ v8f wmma4(v2f a, v2f b, v8f c) {
    // D(16x16,f32) = A(16x4,f32) x B(4x16,f32) + C   — exact fp32 WMMA
    return __builtin_amdgcn_wmma_f32_16x16x4_f32(
        /*neg_a=*/false, a, /*neg_b=*/false, b,
        /*c_mod=*/(short)0, c, /*reuse_a=*/false, /*reuse_b=*/false);
}

static __device__ __forceinline__ void atomic_max_f32_dev(float* p, float v) {
    asm volatile("global_atomic_max_num_f32 %0, %1, off scope:SCOPE_DEV"
                 :: "v"(p), "v"(v) : "memory");
}

static __device__ __forceinline__ v2f ld2(const float* p) {  // p 8B-aligned
    return *(const v2f*)p;
}

// ---------------------------------------------------------------- kernel 1
__global__ __launch_bounds__(256) void k_init_out(float4* out, int n4) {
    int i = blockIdx.x * 256 + threadIdx.x;
    if (i < n4) out[i] = make_float4(-INFINITY, -INFINITY, -INFINITY, -INFINITY);
}

// ---------------------------------------------------------------- kernel 2
// K-pair-interleaved transposed weights so every WMMA B-fragment is ONE
// aligned 8-byte load:
//   fstp[((k/2)*128 + c)*2 + (k&1)] = fs_w[c*128 + k]          (128x128)
//   ptwp[((f/2)*128 + k)*2 + (f&1)] = proj_w[k*512 + f]        (512x128)
__global__ __launch_bounds__(256) void k_transpose(const float* fs_w, const float* proj_w,
                                                   float* fstp, float* ptwp) {
    int t = blockIdx.x * 256 + threadIdx.x;
    if (t < 128 * 128) {
        int r = t & 1, pc = t >> 1;
        int c = pc & 127, p = pc >> 7;
        fstp[t] = fs_w[c * 128 + (2 * p + r)];
    } else if (t < 128 * 128 + 512 * 128) {
        int t2 = t - 128 * 128;
        int r = t2 & 1, pc = t2 >> 1;
        int c = pc & 127, p = pc >> 7;
        ptwp[t2] = proj_w[c * 512 + (2 * p + r)];
    }
}

// ---------------------------------------------------------------- kernel 3
// z[m, :] = h[m, :] @ fs_w^T + fs_b ; M = N*L rows, 16 rows / workgroup,
// 8 waves each own one 16-wide N-tile, K loop via f32 WMMA.
__global__ __launch_bounds__(256) void k_zgemm(const float* __restrict__ h,
                                               const float* __restrict__ fstp,
                                               const float* __restrict__ fs_b,
                                               float* __restrict__ z) {
    __shared__ float lh[16 * 128];
    int t = threadIdx.x;
    size_t rowbase = (size_t)blockIdx.x * 16;
    const float4* h4 = (const float4*)(h + rowbase * 128);
    float4* l4 = (float4*)lh;
    l4[t]       = h4[t];
    l4[t + 256] = h4[t + 256];
    __syncthreads();

    int w = t >> 5, lane = t & 31, hi = lane >> 4, lo = lane & 15;
    int col = w * 16 + lo;
    float bias = fs_b[col];
    v8f c = {bias, bias, bias, bias, bias, bias, bias, bias};
    const float* A = lh + lo * 128;
    for (int kk = 0; kk < 128; kk += 4) {
        int f0 = kk + hi * 2;                       // always even
        v2f a = ld2(A + f0);
        v2f b = ld2(fstp + ((f0 >> 1) * 128 + col) * 2);
        c = wmma4(a, b, c);
    }
    for (int r = 0; r < 8; ++r)
        z[(rowbase + r + hi * 8) * 128 + col] = c[r];
}

// ---------------------------------------------------------------- kernel 4
// ax[m] = wx . z[m,:], ay[m] = wy . z[m,:]  (one wave per row)
__global__ __launch_bounds__(256) void k_rowdots(const float* __restrict__ z,
                                                 const float* __restrict__ wx,
                                                 const float* __restrict__ wy,
                                                 float* __restrict__ ax,
                                                 float* __restrict__ ay) {
    int w = threadIdx.x >> 5, lane = threadIdx.x & 31;
    size_t row = (size_t)blockIdx.x * 8 + w;
    const float* zr = z + row * 128;
    float sx = 0.f, sy = 0.f;
    for (int c = lane; c < 128; c += 32) {
        float v = zr[c];
        sx += v * wx[c];
        sy += v * wy[c];
    }
    for (int off = 16; off >= 1; off >>= 1) {
        sx += __shfl_xor(sx, off, 32);
        sy += __shfl_xor(sy, off, 32);
    }
    if (lane == 0) { ax[row] = sx; ay[row] = sy; }
}

// ---------------------------------------------------------------- kernel 5
// One workgroup (8 waves) per edge. All matrix math via f32 WMMA.
__global__ __launch_bounds__(256) void k_edge(const float* __restrict__ z,
                                              const float* __restrict__ ax,
                                              const float* __restrict__ ay,
                                              const int* __restrict__ lvec,
                                              const int* __restrict__ src_idx,
                                              const int* __restrict__ dst_idx,
                                              const float* __restrict__ wxy,
                                              const float* __restrict__ att_bias,
                                              const float* __restrict__ ptwp,
                                              const float* __restrict__ proj_b,
                                              float* __restrict__ out) {
    __shared__ float s_sf[32 * 128];   // z[src]
    __shared__ float s_sfx[32 * 128];  // z[src] * wxy
    __shared__ float s_df[32 * 128];   // z[dst]
    __shared__ float s_c2q[32 * 128];
    __shared__ float s_S[32 * 32];     // S then P (softmax) in place
    __shared__ float s_M[32];          // rowmax(S)
    __shared__ float s_b[32];          // softmax(rowmax)
    __shared__ float s_q2c[128];
    __shared__ float s_axs[32];
    __shared__ float s_ayd[32];

    int e = blockIdx.x;
    int src = src_idx[e], dst = dst_idx[e];
    int ls = lvec[src], ld = lvec[dst];
    float bias = att_bias[0];
    int t = threadIdx.x;
    int w = t >> 5, lane = t & 31, hi = lane >> 4, lo = lane & 15;

    // ---- stage sf / sfx / df into LDS (z is L2-resident)
    {
        const float4* zs4 = (const float4*)(z + (size_t)src * 4096);
        const float4* zd4 = (const float4*)(z + (size_t)dst * 4096);
        const float4* wxy4 = (const float4*)wxy;
        float4* sf4 = (float4*)s_sf;
        float4* sfx4 = (float4*)s_sfx;
        float4* df4 = (float4*)s_df;
        for (int v = 0; v < 4; ++v) {
            int q4 = t * 4 + v;               // 0..1023
            float4 a = zs4[q4];
            float4 m = wxy4[q4 & 31];         // column chunk (q4*4)&127
            sf4[q4] = a;
            sfx4[q4] = make_float4(a.x * m.x, a.y * m.y, a.z * m.z, a.w * m.w);
            df4[q4] = zd4[q4];
        }
        if (t < 32) { s_axs[t] = ax[src * 32 + t]; s_ayd[t] = ay[dst * 32 + t]; }
    }
    __syncthreads();

    // ---- S = (sf .* wxy) @ df^T  + ax_i + ay_j + bias, masked  (waves 0..3)
    if (w < 4) {
        int ti = w >> 1, tj = w & 1;
        v8f c = {};
        const float* A = s_sfx + (ti * 16 + lo) * 128;
        const float* B = s_df + (tj * 16 + lo) * 128;   // B[k,j] = df[j,k]
        for (int kk = 0; kk < 128; kk += 4) {
            int f0 = kk + hi * 2;
            c = wmma4(ld2(A + f0), ld2(B + f0), c);
        }
        for (int r = 0; r < 8; ++r) {
            int i = ti * 16 + r + hi * 8;
            int j = tj * 16 + lo;
            float v = c[r] + s_axs[i] + s_ayd[j] + bias;
            if (!((i < ls) && (j < ld))) v = MIN_VAL;
            s_S[i * 32 + j] = v;
        }
    }
    __syncthreads();

    // ---- row softmax (P in place) + rowmax M ; each wave owns 4 rows
    for (int rr = 0; rr < 4; ++rr) {
        int i = w * 4 + rr;
        float s = s_S[i * 32 + lane];
        float m = s;
        for (int off = 16; off >= 1; off >>= 1) m = fmaxf(m, __shfl_xor(m, off, 32));
        float p = __expf(s - m);
        float sum = p;
        for (int off = 16; off >= 1; off >>= 1) sum += __shfl_xor(sum, off, 32);
        s_S[i * 32 + lane] = p / sum;
        if (lane == 0) s_M[i] = m;
    }
    __syncthreads();

    // ---- b = softmax(M) (wave 0)
    if (w == 0) {
        float m = s_M[lane];
        float mx = m;
        for (int off = 16; off >= 1; off >>= 1) mx = fmaxf(mx, __shfl_xor(mx, off, 32));
        float p = __expf(m - mx);
        float sum = p;
        for (int off = 16; off >= 1; off >>= 1) sum += __shfl_xor(sum, off, 32);
        s_b[lane] = p / sum;
    }
    __syncthreads();

    // ---- q2c[d] = sum_i b[i] * sf[i,d]
    if (t < 128) {
        float acc = 0.f;
        for (int i = 0; i < 32; ++i) acc += s_b[i] * s_sf[i * 128 + t];
        s_q2c[t] = acc;
    }
    __syncthreads();

    // ---- c2q = P @ df   (16 tiles over 8 waves)
    for (int tt = w; tt < 16; tt += 8) {
        int mi = tt & 1, nj = tt >> 1;
        v8f c = {};
        const float* A = s_S + (mi * 16 + lo) * 32;
        for (int kk = 0; kk < 32; kk += 4) {
            int f0 = kk + hi * 2;
            v2f a = ld2(A + f0);
            v2f b = {s_df[f0 * 128 + nj * 16 + lo], s_df[(f0 + 1) * 128 + nj * 16 + lo]};
            c = wmma4(a, b, c);
        }
        for (int r = 0; r < 8; ++r)
            s_c2q[(mi * 16 + r + hi * 8) * 128 + nj * 16 + lo] = c[r];
    }
    __syncthreads();

    // ---- proj: e = sf@W1^T + c2q@W2^T + (sf.*c2q)@W3^T + (sf.*q2c)@W4^T + pb
    //      then segment-max via device-scope f32 atomic max
    for (int tt = w; tt < 16; tt += 8) {
        int mi = tt & 1, nj = tt >> 1;
        int ncol = nj * 16 + lo;
        float pb = proj_b[ncol];
        v8f c = {pb, pb, pb, pb, pb, pb, pb, pb};
        const float* Asf = s_sf + (mi * 16 + lo) * 128;
        const float* Acq = s_c2q + (mi * 16 + lo) * 128;
        for (int blk = 0; blk < 4; ++blk) {
            // pair-interleaved W^T block: one 8B load per B-fragment
            const float* Bp = ptwp + (size_t)blk * (128 * 128) + ncol * 2;
            for (int kk = 0; kk < 128; kk += 4) {
                int f0 = kk + hi * 2;                 // even
                v2f a;
                if (blk == 0)      a = ld2(Asf + f0);
                else if (blk == 1) a = ld2(Acq + f0);
                else if (blk == 2) a = ld2(Asf + f0) * ld2(Acq + f0);
                else               a = ld2(Asf + f0) * ld2(s_q2c + f0);
                v2f b = ld2(Bp + (f0 >> 1) * 256);
                c = wmma4(a, b, c);
            }
        }
        for (int r = 0; r < 8; ++r) {
            int i = mi * 16 + r + hi * 8;
            float* p = out + (size_t)dst * 4096 + i * 128 + ncol;
            atomic_max_f32_dev(p, c[r]);
        }
    }
}

// ---------------------------------------------------------------- launch
extern "C" void kernel_launch(void* const* d_in, const int* in_sizes, int n_in,
                              void* d_out, int out_size, void* d_ws, size_t ws_size,
                              hipStream_t stream) {
    const float* h       = (const float*)d_in[0];
    const int*   l       = (const int*)d_in[1];
    const int*   src_idx = (const int*)d_in[2];
    const int*   dst_idx = (const int*)d_in[3];
    const float* fs_w    = (const float*)d_in[4];
    const float* fs_b    = (const float*)d_in[5];
    const float* wx      = (const float*)d_in[6];
    const float* wy      = (const float*)d_in[7];
    const float* wxy     = (const float*)d_in[8];
    const float* attb    = (const float*)d_in[9];
    const float* proj_w  = (const float*)d_in[10];
    const float* proj_b  = (const float*)d_in[11];
    float* out = (float*)d_out;

    const int E = in_sizes[2];          // 16000
    const int M = NN * LL;              // 128000 rows

    float* ws   = (float*)d_ws;
    float* z    = ws;                   // N*L*D          = 16,384,000 f
    float* ax   = z + (size_t)M * DD;   // N*L            =    128,000 f
    float* ay   = ax + M;               // N*L
    float* fstp = ay + M;               // 128*128  (pair-interleaved)
    float* ptwp = fstp + 128 * 128;     // 512*128  (pair-interleaved)

    // 1) init output to -inf (atomic-max identity)
    int n4 = out_size / 4;
    k_init_out<<<(n4 + 255) / 256, 256, 0, stream>>>((float4*)out, n4);

    // 2) weight transposes (pair-interleaved)
    k_transpose<<<(128 * 128 + 512 * 128 + 255) / 256, 256, 0, stream>>>(fs_w, proj_w, fstp, ptwp);

    // 3) z = h @ fs_w^T + fs_b
    k_zgemm<<<M / 16, 256, 0, stream>>>(h, fstp, fs_b, z);

    // 4) row dots
    k_rowdots<<<M / 8, 256, 0, stream>>>(z, wx, wy, ax, ay);

    // 5) per-edge co-attention + proj + segment-max
    k_edge<<<E, 256, 0, stream>>>(z, ax, ay, l, src_idx, dst_idx,
                                  wxy, attb, ptwp, proj_b, out);
}